// SOMQuantizer_31688268709992
// MI455X (gfx1250) — compile-verified
//
#include <hip/hip_runtime.h>

typedef float v2f __attribute__((ext_vector_type(2)));
typedef float v8f __attribute__((ext_vector_type(8)));

#define SOM_K   256
#define EMB_D   32
#define SPATIAL 32768            // 32*32*32
#define NVOX    (8 * SPATIAL)    // 262144 voxels
#define WAVES_PER_BLOCK 2
#define DIST_PITCH 257           // pad to stagger LDS banks

// ---------------------------------------------------------------------------
// Main kernel: one wave32 handles a tile of 16 consecutive voxels.
//   dist GEMM via v_wmma_f32_16x16x4_f32 (exact fp32; memory-bound problem,
//   so no precision trade needed), dist rows staged in LDS for the SOM
//   neighbor gather, argmin via shfl butterflies, streaming NT stores.
// ---------------------------------------------------------------------------
__global__ __launch_bounds__(64) void som_main_kernel(
    const float* __restrict__ x,     // [8][32][32768]
    const float* __restrict__ W,     // [256][32]
    float* __restrict__ out_q,       // [8][32][32768]  (d_out + 1)
    float* __restrict__ out_enc,     // [N][256]        (d_out + 1 + N*32)
    float* __restrict__ ws)          // [3] atomics: commit, som, counts
{
    __shared__ float s_ww[SOM_K];
    __shared__ float s_dist[WAVES_PER_BLOCK][16][DIST_PITCH];
    __shared__ int   s_enc[WAVES_PER_BLOCK][16];
    __shared__ float s_mind[WAVES_PER_BLOCK][16];

    const int tid  = threadIdx.x;
    const int wv   = tid >> 5;       // wave in block
    const int ln   = tid & 31;       // lane in wave32
    const int half = ln >> 4;        // 0: K=0,1 side  1: K=2,3 side
    const int v    = ln & 15;        // voxel (M) / code (N) slot in half

    // ---- ||W_k||^2 table (cheap, block-redundant, L2-resident W) ----------
    for (int k = tid; k < SOM_K; k += 64) {
        const float* wr = W + k * EMB_D;
        float s = 0.f;
#pragma unroll
        for (int c = 0; c < EMB_D; ++c) { float t = wr[c]; s += t * t; }
        s_ww[k] = s;
    }
    __syncthreads();

    const int tile = blockIdx.x * WAVES_PER_BLOCK + wv;
    const int n0   = tile * 16;          // first voxel of tile
    const int b    = n0 >> 15;           // batch  (SPATIAL = 2^15)
    const int s0   = n0 & (SPATIAL - 1); // spatial offset (tiles never cross b)

    // ---- A fragments: lane owns voxel v, 16 of its 32 channels ------------
    // 16x4 fp32 A layout: half0 lanes hold K=4s,4s+1; half1 lanes K=4s+2,4s+3
    const float* xp = x + (size_t)b * (EMB_D * SPATIAL) + s0 + v;
    float xa[16];
#pragma unroll
    for (int s = 0; s < 8; ++s) {
        const int c0 = 4 * s + 2 * half;
        xa[2 * s]     = xp[(size_t)c0 * SPATIAL];
        xa[2 * s + 1] = xp[(size_t)(c0 + 1) * SPATIAL];
    }

    // ---- ||x_v||^2 : partial per half, combine across halves --------------
    float ps = 0.f;
#pragma unroll
    for (int i = 0; i < 16; ++i) ps += xa[i] * xa[i];
    const float xxv = ps + __shfl_xor(ps, 16, 32);   // lane holds ||x_v||^2
    float xr[8];                                     // ||x_m||^2 for my acc rows
#pragma unroll
    for (int r = 0; r < 8; ++r) xr[r] = __shfl(xxv, r + 8 * half, 32);

    float minv[8];
    int   minc[8];
#pragma unroll
    for (int r = 0; r < 8; ++r) { minv[r] = 3.4e38f; minc[r] = 0; }

    // ---- sweep 16 code tiles: 8 x wmma_f32_16x16x4 each --------------------
    for (int t = 0; t < 16; ++t) {
        v8f acc = {};
        // B fragment mirrors A layout: lane = code t*16+v, channels 4s+2h,+1
        const float* wb = W + ((t * 16 + v) * EMB_D) + 2 * half;
#pragma unroll
        for (int s = 0; s < 8; ++s) {
            v2f a;
            a.x = xa[2 * s];
            a.y = xa[2 * s + 1];
            v2f bf = *(const v2f*)(wb + 4 * s);   // global_load_b64, L0-hot
            acc = __builtin_amdgcn_wmma_f32_16x16x4_f32(
                false, a, false, bf, (short)0, acc, false, false);
        }
        const float wwv  = s_ww[t * 16 + v];
        const int   code = t * 16 + v;
#pragma unroll
        for (int r = 0; r < 8; ++r) {
            // C/D layout: VGPR r -> row M = r + 8*half, col N = v
            const float d = xr[r] + wwv - 2.0f * acc[r];
            s_dist[wv][r + 8 * half][code] = d;      // ds_store_b32
            if (d < minv[r]) { minv[r] = d; minc[r] = code; }  // first-min order
        }
    }

    // ---- cross-lane argmin within each 16-lane half (lower code on tie) ---
#pragma unroll
    for (int m = 1; m < 16; m <<= 1) {
#pragma unroll
        for (int r = 0; r < 8; ++r) {
            const float ov = __shfl_xor(minv[r], m, 32);
            const int   oc = __shfl_xor(minc[r], m, 32);
            if (ov < minv[r] || (ov == minv[r] && oc < minc[r])) {
                minv[r] = ov; minc[r] = oc;
            }
        }
    }
    if (v == 0) {   // lane 0 publishes rows 0-7, lane 16 rows 8-15
#pragma unroll
        for (int r = 0; r < 8; ++r) {
            s_enc[wv][r + 8 * half]  = minc[r];
            s_mind[wv][r + 8 * half] = minv[r];
        }
    }
    __syncthreads();

    // ---- SOM neighbor gather + commitment from staged dist rows -----------
    float som = 0.f, cntf = 0.f, com = 0.f;
    if (ln < 16) {
        const int   enc = s_enc[wv][v];
        const float md  = s_mind[wv][v];   // dist[v][enc] == ||x - w_enc||^2
        com  = md;
        som  = md;
        cntf = 1.f;
        const int eh = enc >> 4, ew = enc & 15;
        if (eh > 0)  { som += s_dist[wv][v][enc - 16]; cntf += 1.f; }
        if (eh < 15) { som += s_dist[wv][v][enc + 16]; cntf += 1.f; }
        if (ew > 0)  { som += s_dist[wv][v][enc - 1];  cntf += 1.f; }
        if (ew < 15) { som += s_dist[wv][v][enc + 1];  cntf += 1.f; }
    }
#pragma unroll
    for (int m = 1; m < 32; m <<= 1) {
        som  += __shfl_xor(som,  m, 32);
        cntf += __shfl_xor(cntf, m, 32);
        com  += __shfl_xor(com,  m, 32);
    }
    if (ln == 0) {
        atomicAdd(&ws[0], com);
        atomicAdd(&ws[1], som);
        atomicAdd(&ws[2], cntf);
    }

    // ---- quantized output: out[b][c][s0+v] = W[enc_v][c], NT stores -------
    {
        const int enc = s_enc[wv][v];
        float* op       = out_q + (size_t)b * (EMB_D * SPATIAL) + s0 + v;
        const float* wq = W + enc * EMB_D;
#pragma unroll
        for (int i = 0; i < 16; ++i) {
            const int c = i + 16 * half;   // halves cover channel ranges
            __builtin_nontemporal_store(wq[c], op + (size_t)c * SPATIAL);
        }
    }

    // ---- one-hot encodings: coalesced 32-lane rows, NT streaming ----------
    {
        float* ep = out_enc + (size_t)n0 * SOM_K;
        for (int vv = 0; vv < 16; ++vv) {
            const int enc = s_enc[wv][vv];  // uniform LDS broadcast
#pragma unroll
            for (int it = 0; it < 8; ++it) {
                const int code = it * 32 + ln;   // 32 consecutive dwords/instr
                __builtin_nontemporal_store(code == enc ? 1.0f : 0.0f,
                                            ep + (size_t)vv * SOM_K + code);
            }
        }
    }
}

// ---------------------------------------------------------------------------
__global__ void som_finalize_kernel(const float* __restrict__ ws,
                                    float* __restrict__ loss)
{
    const float commit_mean = ws[0] / (float)((size_t)NVOX * EMB_D);
    const float cnt = ws[2];
    const float soml = (cnt > 0.f) ? (ws[1] / cnt) : 0.f;
    loss[0] = 6.0f * commit_mean + 1.0f * soml;   // ALPHA=6, BETA=1
}

// ---------------------------------------------------------------------------
extern "C" void kernel_launch(void* const* d_in, const int* in_sizes, int n_in,
                              void* d_out, int out_size, void* d_ws, size_t ws_size,
                              hipStream_t stream) {
    (void)in_sizes; (void)n_in; (void)out_size; (void)ws_size;
    const float* x = (const float*)d_in[0];   // [8,32,32,32,32] fp32
    const float* W = (const float*)d_in[1];   // [256,32] fp32
    float* out = (float*)d_out;               // [loss(1) | out(N*32) | enc(N*256)]
    float* ws  = (float*)d_ws;

    hipMemsetAsync(ws, 0, 4 * sizeof(float), stream);   // graph-capturable

    float* out_q   = out + 1;
    float* out_enc = out + 1 + (size_t)NVOX * EMB_D;

    const int tiles  = NVOX / 16;                 // 16384
    const int blocks = tiles / WAVES_PER_BLOCK;   // 8192
    som_main_kernel<<<dim3(blocks), dim3(64), 0, stream>>>(x, W, out_q, out_enc, ws);
    som_finalize_kernel<<<dim3(1), dim3(1), 0, stream>>>(ws, out);
}